// image_prj_39926015984313
// MI455X (gfx1250) — compile-verified
//
#include <hip/hip_runtime.h>
#include <math.h>

// ---- problem constants (match reference) ----
#define IMG_N     2048
#define HI_N      1024            // dwt-high output: 1024x1024
#define PAD_N     1449            // padded square side
#define PAD_OFF   212             // placement offset of high image inside padded
#define CENTER    724.0f          // PAD_N // 2
#define OUT_N     1448            // out_h == out_w == PAD_N - 1
#define N_ANG     64
#define N_SEG     4               // y-range split: 4 segments of 362 rows
#define SEG_ROWS  362             // 1448 / 4
#define LINE_ELEMS (N_ANG * OUT_N)        // 92672
#define PADDED_ELEMS (PAD_N * PAD_N)      // 2099601

// ws layout (in floats)
#define WS_PADDED   0
#define WS_PARTIAL  (PADDED_ELEMS)                          // N_SEG * LINE_ELEMS
#define WS_LINES    (WS_PARTIAL + N_SEG * LINE_ELEMS)       // LINE_ELEMS
#define WS_BLKMAX   (WS_LINES + LINE_ELEMS)                 // 362
#define WS_MAXV     (WS_BLKMAX + 512)                       // 1

typedef __attribute__((ext_vector_type(2))) float v2f;
typedef __attribute__((ext_vector_type(8))) float v8f;

// ---- kernel A: zero the padded buffer ----
__global__ void k_zero(float* __restrict__ p, int n) {
    int i = blockIdx.x * blockDim.x + threadIdx.x;
    if (i < n) p[i] = 0.0f;
}

// ---- kernel B: stride-2 2x2 high-pass conv, write into padded at offset ----
__global__ void k_dwt(const float* __restrict__ img, float* __restrict__ padded) {
    int idx = blockIdx.x * blockDim.x + threadIdx.x;
    if (idx >= HI_N * HI_N) return;
    int i = idx >> 10;          // row (HI_N == 1024)
    int j = idx & 1023;         // col
    const float* r0 = img + (size_t)(2 * i) * IMG_N + 2 * j;
    const float* r1 = r0 + IMG_N;
    // cross-correlation with HIGH_F (no kernel flip in XLA conv)
    float v = fmaf(-0.1384f, r0[0],
              fmaf( 0.7243f, r0[1],
              fmaf(-0.6038f, r1[0],
                    0.1601f * r1[1])));
    padded[(size_t)(PAD_OFF + i) * PAD_N + (PAD_OFF + j)] = v;
}

// ---- kernel C: rotate + bilinear sample + WMMA column-sum reduction ----
// grid: (12 x-blocks, 64 angles, 4 y-segments), block = 256 threads (8 waves)
// Each wave owns 16 consecutive x columns. Per iteration a wave produces a
// 4x16 f32 B-matrix of samples (2 per lane); A = ones(16x4) so
// D = A*B + C accumulates the 16 column sums in every row of D (f32 end-to-end).
__global__ void k_rotsum(const float* __restrict__ padded, float* __restrict__ partial) {
    const int lane  = threadIdx.x & 31;
    const int wave  = threadIdx.x >> 5;
    const int n     = lane & 15;       // column within wave's tile
    const int ysub  = lane >> 4;       // 0 or 1
    const int x     = blockIdx.x * 128 + wave * 16 + n;
    const int ang   = blockIdx.y;
    const int seg   = blockIdx.z;
    const int ystart = seg * SEG_ROWS;

    const float a  = (float)ang;
    const float c  = cosf(a);
    const float s  = sinf(a);
    const float Cx = -CENTER * (c + s - 1.0f);
    const float Cy = -CENTER * (c - s - 1.0f);
    const float xf = (float)x;
    const float bx = fmaf(c, xf, Cx);      // x-dependent part of x_in
    const float by = fmaf(-s, xf, Cy);     // x-dependent part of y_in

    v8f acc = {};                          // C/D accumulator (16x16 f32)
    const v2f ones = {1.0f, 1.0f};         // A = ones 16x4

    // 362 rows per segment -> 91 iterations of 4 rows (last iter masks 2 rows)
    for (int it = 0; it < 91; ++it) {
        v2f b;
        #pragma unroll
        for (int j = 0; j < 2; ++j) {
            const int yrel = 4 * it + ysub + 2 * j;
            const float yf = (float)(ystart + yrel);
            const float x_in = fmaf(s, yf, bx);
            const float y_in = fmaf(c, yf, by);
            const float fx = floorf(x_in);
            const float fy = floorf(y_in);
            const float dx = x_in - fx;
            const float dy = y_in - fy;
            int x0 = (int)fx % PAD_N; if (x0 < 0) x0 += PAD_N;
            int y0 = (int)fy % PAD_N; if (y0 < 0) y0 += PAD_N;
            int x1 = x0 + 1; if (x1 == PAD_N) x1 = 0;
            int y1 = y0 + 1; if (y1 == PAD_N) y1 = 0;
            const float v00 = padded[(size_t)y0 * PAD_N + x0];
            const float v01 = padded[(size_t)y0 * PAD_N + x1];
            const float v10 = padded[(size_t)y1 * PAD_N + x0];
            const float v11 = padded[(size_t)y1 * PAD_N + x1];
            const float omdx = 1.0f - dx, omdy = 1.0f - dy;
            float val = v00 * omdy * omdx + v01 * omdy * dx
                      + v10 * dy * omdx   + v11 * dy * dx;
            // mask tail rows by zeroing values (keeps EXEC all-1s for WMMA)
            b[j] = (yrel < SEG_ROWS) ? val : 0.0f;
        }
        // D = ones(16x4) x B(4x16) + C  => every D row = column sums of B
        acc = __builtin_amdgcn_wmma_f32_16x16x4_f32(
            /*neg_a=*/false, ones, /*neg_b=*/false, b,
            /*c_mod=*/(short)0, acc, /*reuse_a=*/false, /*reuse_b=*/false);
    }

    // D VGPR0 on lanes 0..15 holds (M=0, N=lane) = column sum for column n
    if (lane < 16 && x < OUT_N)
        partial[((size_t)seg * N_ANG + ang) * OUT_N + x] = acc[0];
}

// ---- kernel D: sum 4 segments -> lines; per-block max (deterministic tree) ----
__global__ void k_lines_max(const float* __restrict__ partial,
                            float* __restrict__ lines,
                            float* __restrict__ blkmax) {
    __shared__ float sm[256];
    const int idx = blockIdx.x * 256 + threadIdx.x;   // 362 blocks * 256 = 92672
    float v = partial[idx]
            + partial[idx + 1 * LINE_ELEMS]
            + partial[idx + 2 * LINE_ELEMS]
            + partial[idx + 3 * LINE_ELEMS];
    lines[idx] = v;
    sm[threadIdx.x] = v;
    __syncthreads();
    for (int off = 128; off > 0; off >>= 1) {
        if (threadIdx.x < off)
            sm[threadIdx.x] = fmaxf(sm[threadIdx.x], sm[threadIdx.x + off]);
        __syncthreads();
    }
    if (threadIdx.x == 0) blkmax[blockIdx.x] = sm[0];
}

// ---- kernel E: reduce 362 block maxes -> scalar max ----
__global__ void k_final_max(const float* __restrict__ blkmax, float* __restrict__ maxv) {
    __shared__ float sm[512];
    float m = -INFINITY;
    for (int i = threadIdx.x; i < 362; i += 512) m = fmaxf(m, blkmax[i]);
    sm[threadIdx.x] = m;
    __syncthreads();
    for (int off = 256; off > 0; off >>= 1) {
        if (threadIdx.x < off)
            sm[threadIdx.x] = fmaxf(sm[threadIdx.x], sm[threadIdx.x + off]);
        __syncthreads();
    }
    if (threadIdx.x == 0) maxv[0] = sm[0];
}

// ---- kernel F: A = lines.T / max ----
__global__ void k_norm(const float* __restrict__ lines, const float* __restrict__ maxv,
                       float* __restrict__ out) {
    int i = blockIdx.x * blockDim.x + threadIdx.x;
    if (i >= LINE_ELEMS) return;
    int a = i & 63;       // N_ANG == 64
    int x = i >> 6;
    out[i] = lines[(size_t)a * OUT_N + x] / maxv[0];
}

extern "C" void kernel_launch(void* const* d_in, const int* in_sizes, int n_in,
                              void* d_out, int out_size, void* d_ws, size_t ws_size,
                              hipStream_t stream) {
    const float* image = (const float*)d_in[0];
    float* ws = (float*)d_ws;
    float* padded  = ws + WS_PADDED;
    float* partial = ws + WS_PARTIAL;
    float* lines   = ws + WS_LINES;
    float* blkmax  = ws + WS_BLKMAX;
    float* maxv    = ws + WS_MAXV;
    float* out     = (float*)d_out;

    // A: zero padded buffer
    k_zero<<<(PADDED_ELEMS + 255) / 256, 256, 0, stream>>>(padded, PADDED_ELEMS);
    // B: DWT high-pass into padded
    k_dwt<<<(HI_N * HI_N + 255) / 256, 256, 0, stream>>>(image, padded);
    // C: rotate + sample + WMMA column sums
    dim3 grid(12, N_ANG, N_SEG);   // 12*128 = 1536 >= 1448 columns
    k_rotsum<<<grid, 256, 0, stream>>>(padded, partial);
    // D: segment sums + block maxes
    k_lines_max<<<LINE_ELEMS / 256, 256, 0, stream>>>(partial, lines, blkmax);
    // E: final max
    k_final_max<<<1, 512, 0, stream>>>(blkmax, maxv);
    // F: transpose + normalize
    k_norm<<<(LINE_ELEMS + 255) / 256, 256, 0, stream>>>(lines, maxv, out);
}